// LTFGW_35510789604051
// MI455X (gfx1250) — compile-verified
//
#include <hip/hip_runtime.h>

// LTFGW fused kernel for gfx1250 (MI455X).
// One 256-thread block (8 wave32) per graph node. FP32 WMMA 16x16x4 for the
// neighbor-feature x template-feature GEMM; fully LDS-resident FGW/Sinkhorn.

#define N_NODES 20000
#define DEG     16
#define NSUB    17          // ego-subgraph size n = DEG+1
#define T_N     16          // templates
#define M_T     8           // template nodes
#define D_F     64          // feature dim
#define NCOL    (T_N * M_T) // 128 flattened template columns
#define ALPHA   0.5f
#define EPS_R   0.1f
#define INV_EPS 10.0f
#define N_OUTER 4
#define N_INNER 10

typedef float v2f __attribute__((ext_vector_type(2)));
typedef float v8f __attribute__((ext_vector_type(8)));

__global__ __launch_bounds__(256) void ltfgw_fused_kernel(
    const float* __restrict__ x,          // [N_NODES, D_F]
    const int*   __restrict__ edge_index, // [2, N_NODES*DEG]
    const float* __restrict__ tf,         // [T_N, M_T, D_F]
    const float* __restrict__ tm,         // [T_N, M_T, M_T]
    float*       __restrict__ out)        // [N_NODES, T_N]
{
    // ---- LDS (static; ~62.5 KB total) ----
    __shared__ float sTFt  [D_F * NCOL];        // templates transposed: [k][col]
    __shared__ float sFlocT[D_F * NSUB];        // ego features transposed: [k][row]
    __shared__ float sC2   [T_N * M_T * M_T];   // template structure
    __shared__ float sMbase[T_N * NSUB * M_T];  // (1-a)*M + a*constC
    __shared__ float sCost [T_N * NSUB * M_T];  // current linearized cost
    __shared__ float sF    [T_N * NSUB];        // sinkhorn f
    __shared__ float sGv   [T_N * M_T];         // sinkhorn g
    __shared__ float sS  [NCOL], sR0 [NCOL];    // column sums of G (rows 1.., row 0)
    __shared__ float sU  [NCOL], sV  [NCOL];    // S@C2, R0@C2
    __shared__ float sSqT[NCOL], sBvec[NCOL];   // ||tf_col||^2 ; b[t,m]
    __shared__ float sSqF[NSUB];
    __shared__ int   sNeigh[NSUB];
    __shared__ float sPart[NCOL];

    const int tid  = threadIdx.x;
    const int node = blockIdx.x;
    const int lane = tid & 31;
    const int wv   = tid >> 5;                  // wave id 0..7

    const float LOGP = -2.83321334f;            // log(1/17)
    const float LOGQ = -2.07944154f;            // log(1/8)

    // ---- stage: neighbor list, C2, templates (k-major) ----
    if (tid < NSUB)
        sNeigh[tid] = (tid == 0) ? node
                                 : edge_index[(size_t)N_NODES * DEG + node * DEG + (tid - 1)];
    for (int e = tid; e < T_N * M_T * M_T; e += 256) sC2[e] = tm[e];
    for (int e = tid; e < NCOL * D_F; e += 256) {
        int c = e >> 6, k = e & 63;             // tf is [col][k] row-major
        sTFt[k * NCOL + c] = tf[e];
    }
    __syncthreads();
    for (int e = tid; e < NSUB * D_F; e += 256) {
        int r = e >> 6, k = e & 63;
        sFlocT[k * NSUB + r] = x[(size_t)sNeigh[r] * D_F + k];
    }
    __syncthreads();

    // ---- squared norms and b[t,m] = (1/8) sum_k C2[t,m,k]^2 ----
    if (tid < NSUB) {
        float s = 0.f;
        for (int k = 0; k < D_F; ++k) { float v = sFlocT[k * NSUB + tid]; s += v * v; }
        sSqF[tid] = s;
    }
    if (tid < NCOL) {
        float s = 0.f;
        for (int k = 0; k < D_F; ++k) { float v = sTFt[k * NCOL + tid]; s += v * v; }
        sSqT[tid] = s;
        int t = tid >> 3, j = tid & 7;
        float bb = 0.f;
        for (int k = 0; k < M_T; ++k) { float v = sC2[(t * M_T + j) * M_T + k]; bb += v * v; }
        sBvec[tid] = bb * 0.125f;
    }
    __syncthreads();

    // ---- WMMA: neighbors(16x64) @ TF^T(64x128), wave wv owns cols [16wv,16wv+16) ----
    // A frag (16x4 f32): lane m = lane&15 -> neighbor row, K = (lane>>4)*2 + v
    // B frag (4x16 f32): lane n = lane&15 -> column,       K = (lane>>4)*2 + v
    v8f acc = {0.f, 0.f, 0.f, 0.f, 0.f, 0.f, 0.f, 0.f};
    {
        const int half = lane >> 4;
        const int mrow = lane & 15;
        const int col  = wv * 16 + mrow;
        #pragma unroll
        for (int kt = 0; kt < 16; ++kt) {
            const int k0 = kt * 4 + half * 2;
            v2f a, b;
            a.x = sFlocT[(k0    ) * NSUB + (mrow + 1)];
            a.y = sFlocT[(k0 + 1) * NSUB + (mrow + 1)];
            b.x = sTFt  [(k0    ) * NCOL + col];
            b.y = sTFt  [(k0 + 1) * NCOL + col];
            acc = __builtin_amdgcn_wmma_f32_16x16x4_f32(
                false, a, false, b, (short)0, acc, false, false);
        }
    }
    // scatter D-tile: lane l vgpr r holds M = r + 8*(l>>4), N = l&15
    {
        const int half = lane >> 4;
        const int ncol = lane & 15;
        const int col  = wv * 16 + ncol;
        const int t = col >> 3, j = col & 7;
        #pragma unroll
        for (int r = 0; r < 8; ++r) {
            const int i = (r + 8 * half) + 1;   // subgraph rows 1..16
            float mc = sSqF[i] + sSqT[col] - 2.f * acc[r];
            sMbase[(t * NSUB + i) * M_T + j] =
                (1.f - ALPHA) * mc + ALPHA * ((1.f / 17.f) + sBvec[col]);
        }
    }
    // center row (i = 0) scalar dot; also init S, R0 for uniform G = pq^T
    if (tid < NCOL) {
        float d = 0.f;
        for (int k = 0; k < D_F; ++k) d += sFlocT[k * NSUB + 0] * sTFt[k * NCOL + tid];
        int t = tid >> 3, j = tid & 7;
        float mc = sSqF[0] + sSqT[tid] - 2.f * d;
        sMbase[(t * NSUB + 0) * M_T + j] =
            (1.f - ALPHA) * mc + ALPHA * ((16.f / 17.f) + sBvec[tid]);
        sS [tid] = 16.f / 136.f;                // sum_{i>=1} 1/(n*m)
        sR0[tid] = 1.f  / 136.f;
    }
    __syncthreads();

    // ---- outer FGW iterations ----
    for (int outer = 0; outer < N_OUTER; ++outer) {
        // U = S @ C2, V = R0 @ C2 (C2 symmetric); reset g = 0
        if (tid < NCOL) {
            int t = tid >> 3, j = tid & 7;
            float u = 0.f, v = 0.f;
            #pragma unroll
            for (int k = 0; k < M_T; ++k) {
                float c2 = sC2[(t * M_T + j) * M_T + k];
                u += sS [t * M_T + k] * c2;
                v += sR0[t * M_T + k] * c2;
            }
            sU[tid] = u; sV[tid] = v;
            sGv[tid] = 0.f;
        }
        __syncthreads();
        // cost = Mbase - 2*alpha * (i==0 ? U : V)
        for (int e = tid; e < T_N * NSUB * M_T; e += 256) {
            int rr = e >> 3;                    // t*17 + i
            int j  = e & 7;
            int t  = rr / NSUB;
            int i  = rr - t * NSUB;
            float sel = (i == 0) ? sU[t * M_T + j] : sV[t * M_T + j];
            sCost[e] = sMbase[e] - 2.f * ALPHA * sel;
        }
        __syncthreads();

        // ---- log-domain Sinkhorn ----
        for (int it = 0; it < N_INNER; ++it) {
            // f update: 272 rows, LSE over m=8
            for (int rr = tid; rr < T_N * NSUB; rr += 256) {
                int t = rr / NSUB;
                float vv[8], mx = -1e30f;
                #pragma unroll
                for (int j = 0; j < 8; ++j) {
                    float v = (sGv[t * M_T + j] - sCost[rr * M_T + j]) * INV_EPS;
                    vv[j] = v; mx = fmaxf(mx, v);
                }
                float s = 0.f;
                #pragma unroll
                for (int j = 0; j < 8; ++j) s += __expf(vv[j] - mx);
                sF[rr] = EPS_R * (LOGP - (mx + __logf(s)));
            }
            __syncthreads();
            // g update: 128 cols, LSE over n=17
            if (tid < NCOL) {
                int t = tid >> 3, j = tid & 7;
                float vv[NSUB], mx = -1e30f;
                #pragma unroll
                for (int i = 0; i < NSUB; ++i) {
                    float v = (sF[t * NSUB + i] - sCost[(t * NSUB + i) * M_T + j]) * INV_EPS;
                    vv[i] = v; mx = fmaxf(mx, v);
                }
                float s = 0.f;
                #pragma unroll
                for (int i = 0; i < NSUB; ++i) s += __expf(vv[i] - mx);
                sGv[tid] = EPS_R * (LOGQ - (mx + __logf(s)));
            }
            __syncthreads();
        }

        // S, R0 from implicit G = exp((f + g - cost)/eps)
        if (tid < NCOL) {
            int t = tid >> 3, j = tid & 7;
            float g = sGv[tid];
            float r0 = __expf((sF[t * NSUB + 0] + g - sCost[(t * NSUB + 0) * M_T + j]) * INV_EPS);
            float s = 0.f;
            #pragma unroll
            for (int i = 1; i < NSUB; ++i)
                s += __expf((sF[t * NSUB + i] + g - sCost[(t * NSUB + i) * M_T + j]) * INV_EPS);
            sR0[tid] = r0; sS[tid] = s;
        }
        __syncthreads();
    }

    // ---- final FGW value: sum(G * ((1-a)M + a*tens(G_final))) ----
    if (tid < NCOL) {                           // U,V from FINAL G
        int t = tid >> 3, j = tid & 7;
        float u = 0.f, v = 0.f;
        #pragma unroll
        for (int k = 0; k < M_T; ++k) {
            float c2 = sC2[(t * M_T + j) * M_T + k];
            u += sS [t * M_T + k] * c2;
            v += sR0[t * M_T + k] * c2;
        }
        sU[tid] = u; sV[tid] = v;
    }
    __syncthreads();
    if (tid < NCOL) {
        int t = tid >> 3, j = tid & 7;
        float g = sGv[tid];
        float a = 0.f;
        #pragma unroll
        for (int i = 0; i < NSUB; ++i) {
            int e = (t * NSUB + i) * M_T + j;
            float G  = __expf((sF[t * NSUB + i] + g - sCost[e]) * INV_EPS);
            float cf = sMbase[e] - 2.f * ALPHA * ((i == 0) ? sU[tid] : sV[tid]);
            a += G * cf;
        }
        sPart[tid] = a;
    }
    __syncthreads();
    if (tid < T_N) {
        float s = 0.f;
        #pragma unroll
        for (int j = 0; j < 8; ++j) s += sPart[tid * M_T + j];
        out[(size_t)node * T_N + tid] = s;
    }
}

extern "C" void kernel_launch(void* const* d_in, const int* in_sizes, int n_in,
                              void* d_out, int out_size, void* d_ws, size_t ws_size,
                              hipStream_t stream) {
    (void)in_sizes; (void)n_in; (void)out_size; (void)d_ws; (void)ws_size;
    const float* x  = (const float*)d_in[0];
    const int*   ei = (const int*)  d_in[1];
    const float* tf = (const float*)d_in[2];
    const float* tm = (const float*)d_in[3];
    float* out = (float*)d_out;
    ltfgw_fused_kernel<<<N_NODES, 256, 0, stream>>>(x, ei, tf, tm, out);
}